// BahdanauAttention_87076166959132
// MI455X (gfx1250) — compile-verified
//
#include <hip/hip_runtime.h>
#include <hip/hip_bf16.h>

typedef __attribute__((ext_vector_type(2))) float v2f;
typedef __attribute__((ext_vector_type(4))) float v4f;
typedef __attribute__((ext_vector_type(8))) float v8f;

// Problem dims (fixed by the reference)
#define SEQ   2048
#define BATCH 64
#define ENC_H 1024
#define DEC_H 1024
#define ATT   512

// ---------------------------------------------------------------------------
// K1: dec_att (64 x 512) = dec_out (64 x 1024) @ W_dec^T (W_dec is 512 x 1024)
// One wave per 16x16 tile of C; K-loop in steps of 4 via v_wmma_f32_16x16x4_f32.
// A 16x4 f32 layout: lanes 0-15 hold K=0 (v0), K=1 (v1); lanes 16-31 K=2,3.
// B 4x16 f32 layout mirrors A with N = lane%16.
// C/D: VGPR i -> M = i (lanes 0-15) / M = i+8 (lanes 16-31), N = lane%16.
// ---------------------------------------------------------------------------
__global__ __launch_bounds__(32) void gemm_dec_att_wmma(
    const float* __restrict__ A,   // dec_out (64 x 1024) row-major
    const float* __restrict__ W,   // W_dec   (512 x 1024) row-major
    float* __restrict__ C)         // dec_att (64 x 512)  row-major
{
    const int tile = blockIdx.x;       // 128 tiles: 4 (M) x 32 (N)
    const int mt   = tile & 3;
    const int nt   = tile >> 2;
    const int lane = threadIdx.x;      // wave32
    const int r    = lane & 15;
    const int kOff = (lane >> 4) * 2;  // 0 or 2

    const float* Arow = A + (size_t)(mt * 16 + r) * DEC_H;
    const float* Wrow = W + (size_t)(nt * 16 + r) * DEC_H;

    v8f c = {};
    for (int k0 = 0; k0 < DEC_H; k0 += 4) {
        v2f a = *(const v2f*)(Arow + k0 + kOff);
        v2f b = *(const v2f*)(Wrow + k0 + kOff);
        // 8 args: (neg_a, A, neg_b, B, c_mod, C, reuse_a, reuse_b)
        c = __builtin_amdgcn_wmma_f32_16x16x4_f32(false, a, false, b,
                                                  (short)0, c, false, false);
    }

    const int row0 = mt * 16 + (lane >> 4) * 8;
    const int col  = nt * 16 + r;
#pragma unroll
    for (int i = 0; i < 8; ++i)
        C[(size_t)(row0 + i) * ATT + col] = c[i];
}

// ---------------------------------------------------------------------------
// K2: scores[b,s] = sum_a tanh(enc_att[s,b,a] + dec_att[b,a]) * att_v[a]
// One wave per (s,b). enc_att streamed once (268 MB) -> non-temporal loads.
// dec_att / att_v are tiny and hot in L2.
// ---------------------------------------------------------------------------
__global__ __launch_bounds__(256) void scores_kernel(
    const float* __restrict__ enc_att,  // (SEQ, BATCH, ATT)
    const float* __restrict__ dec_att,  // (BATCH, ATT)
    const float* __restrict__ att_v,    // (ATT)
    float* __restrict__ scores)         // (BATCH, SEQ)
{
    const int wave = (blockIdx.x * blockDim.x + threadIdx.x) >> 5;
    const int lane = threadIdx.x & 31;
    const int s = wave >> 6;       // SEQ index
    const int b = wave & 63;       // batch index

    const float* e = enc_att + ((size_t)s * BATCH + b) * ATT;
    const float* d = dec_att + (size_t)b * ATT;

    float acc = 0.f;
#pragma unroll
    for (int it = 0; it < 4; ++it) {
        const int a0 = it * 128 + lane * 4;              // 512B/wave/iter, coalesced
        v4f ev = __builtin_nontemporal_load((const v4f*)(e + a0));
        v4f dv = *(const v4f*)(d + a0);
        v4f vv = *(const v4f*)(att_v + a0);
        acc += tanhf(ev.x + dv.x) * vv.x;
        acc += tanhf(ev.y + dv.y) * vv.y;
        acc += tanhf(ev.z + dv.z) * vv.z;
        acc += tanhf(ev.w + dv.w) * vv.w;
    }
#pragma unroll
    for (int off = 16; off; off >>= 1)
        acc += __shfl_xor(acc, off, 32);
    if (lane == 0)
        scores[(size_t)b * SEQ + s] = acc;
}

// ---------------------------------------------------------------------------
// K3: softmax over seq per batch row; writes weights straight into d_out slice
// ---------------------------------------------------------------------------
__global__ __launch_bounds__(256) void softmax_kernel(
    const float* __restrict__ scores,   // (BATCH, SEQ)
    float* __restrict__ weights)        // (BATCH, SEQ)
{
    __shared__ float red[8];
    const int b = blockIdx.x;
    const int t = threadIdx.x;
    const int lane = t & 31, w = t >> 5;
    const float* row = scores + (size_t)b * SEQ;

    float m = -3.4e38f;
    for (int s = t; s < SEQ; s += 256) m = fmaxf(m, row[s]);
#pragma unroll
    for (int off = 16; off; off >>= 1) m = fmaxf(m, __shfl_xor(m, off, 32));
    if (lane == 0) red[w] = m;
    __syncthreads();
    m = red[0];
#pragma unroll
    for (int i = 1; i < 8; ++i) m = fmaxf(m, red[i]);
    __syncthreads();

    float sum = 0.f;
    for (int s = t; s < SEQ; s += 256) sum += __expf(row[s] - m);
#pragma unroll
    for (int off = 16; off; off >>= 1) sum += __shfl_xor(sum, off, 32);
    if (lane == 0) red[w] = sum;
    __syncthreads();
    float tot = 0.f;
#pragma unroll
    for (int i = 0; i < 8; ++i) tot += red[i];

    const float inv = 1.f / tot;
    for (int s = t; s < SEQ; s += 256)
        weights[(size_t)b * SEQ + s] = __expf(row[s] - m) * inv;
}

// ---------------------------------------------------------------------------
// K4: partial[c,b,h] = sum_{s in chunk c} weights[b,s] * enc_outs[s,b,h]
// 8 seq-chunks x 64 batches x 4 h-blocks = 2048 blocks -> fills the WGPs.
// enc_outs streamed once (537 MB) -> non-temporal loads. Weights chunk in LDS.
// ---------------------------------------------------------------------------
#define NCHUNK 8
#define SCHUNK (SEQ / NCHUNK)   // 256

__global__ __launch_bounds__(256) void context_partial_kernel(
    const float* __restrict__ enc_outs, // (SEQ, BATCH, ENC_H)
    const float* __restrict__ weights,  // (BATCH, SEQ)
    float* __restrict__ partial)        // (NCHUNK, BATCH, ENC_H)
{
    __shared__ float sw[SCHUNK];
    const int h  = blockIdx.x * 256 + threadIdx.x;  // ENC_H/256 = 4 blocks
    const int b  = blockIdx.y;
    const int c  = blockIdx.z;
    const int s0 = c * SCHUNK;

    sw[threadIdx.x] = weights[(size_t)b * SEQ + s0 + threadIdx.x];
    __syncthreads();

    const float* e = enc_outs + ((size_t)s0 * BATCH + b) * ENC_H + h;
    float acc = 0.f;
#pragma unroll 4
    for (int i = 0; i < SCHUNK; ++i)
        acc += sw[i] * __builtin_nontemporal_load(e + (size_t)i * BATCH * ENC_H);

    partial[((size_t)c * BATCH + b) * ENC_H + h] = acc;
}

// ---------------------------------------------------------------------------
// K5: context[b,h] = sum_c partial[c,b,h]
// ---------------------------------------------------------------------------
__global__ __launch_bounds__(256) void context_reduce_kernel(
    const float* __restrict__ partial,  // (NCHUNK, BATCH, ENC_H)
    float* __restrict__ context)        // (BATCH, ENC_H)
{
    const int i = blockIdx.x * 256 + threadIdx.x;   // 0 .. 65535
    float acc = 0.f;
#pragma unroll
    for (int c = 0; c < NCHUNK; ++c)
        acc += partial[(size_t)c * BATCH * ENC_H + i];
    context[i] = acc;
}

// ---------------------------------------------------------------------------
extern "C" void kernel_launch(void* const* d_in, const int* in_sizes, int n_in,
                              void* d_out, int out_size, void* d_ws, size_t ws_size,
                              hipStream_t stream) {
    const float* dec_out  = (const float*)d_in[0];  // (B, DEC_H)
    const float* enc_outs = (const float*)d_in[1];  // (SEQ, B, ENC_H)
    const float* enc_att  = (const float*)d_in[2];  // (SEQ, B, ATT)
    const float* W_dec    = (const float*)d_in[3];  // (ATT, DEC_H)
    const float* att_v    = (const float*)d_in[4];  // (ATT)

    float* out     = (float*)d_out;
    float* context = out;                       // BATCH * ENC_H = 65536 floats
    float* weights = out + BATCH * ENC_H;       // BATCH * SEQ   = 131072 floats

    float* ws      = (float*)d_ws;
    float* dec_att = ws;                                  // 32768 floats
    float* scores  = ws + BATCH * ATT;                    // 131072 floats
    float* partial = ws + BATCH * ATT + BATCH * SEQ;      // 8*65536 floats

    // K1: WMMA GEMM (4 M-tiles x 32 N-tiles = 128 waves)
    gemm_dec_att_wmma<<<128, 32, 0, stream>>>(dec_out, W_dec, dec_att);

    // K2: 131072 (s,b) waves, 8 waves per 256-thread block
    scores_kernel<<<(SEQ * BATCH) / 8, 256, 0, stream>>>(enc_att, dec_att, att_v, scores);

    // K3: one block per batch row
    softmax_kernel<<<BATCH, 256, 0, stream>>>(scores, weights);

    // K4: (h-blocks, batch, seq-chunk)
    dim3 g4(ENC_H / 256, BATCH, NCHUNK);
    context_partial_kernel<<<g4, 256, 0, stream>>>(enc_outs, weights, partial);

    // K5: final reduction into context
    context_reduce_kernel<<<(BATCH * ENC_H) / 256, 256, 0, stream>>>(partial, context);
}